// SemiMarkovConditionalRandomField_29875792511300
// MI455X (gfx1250) — compile-verified
//
#include <hip/hip_runtime.h>

typedef __attribute__((ext_vector_type(2))) float v2f;
typedef __attribute__((ext_vector_type(8))) float v8f;

#define NEG_INF (-1e30f)

// Problem constants (match reference)
#define Bsz 16
#define Ssz 512
#define Wsz 16
#define Tsz 64

// ---------------------------------------------------------------------------
// Kernel 1: numerator.  One block per batch element b.
// numerator[b] = sum_{s,w} inputs[b,s,w,tag] * (tag != 0) * mask[b,s]
// Writes d_out[1 + b].
// ---------------------------------------------------------------------------
__global__ __launch_bounds__(256) void crf_num(const float* __restrict__ inp,
                                               const int* __restrict__ tags,
                                               const int* __restrict__ mask,
                                               float* __restrict__ out) {
  const int b = blockIdx.x;
  float acc = 0.0f;
  for (int i = threadIdx.x; i < Ssz * Wsz; i += 256) {
    const int idx = b * (Ssz * Wsz) + i;     // (b,s,w) flat
    const int tag = tags[idx];
    const int s = i >> 4;
    if (tag != 0 && mask[b * Ssz + s] != 0) {
      acc += inp[(size_t)idx * Tsz + tag];
    }
  }
#pragma unroll
  for (int d = 1; d < 32; d <<= 1) acc += __shfl_xor(acc, d);
  __shared__ float red[8];
  if ((threadIdx.x & 31) == 0) red[threadIdx.x >> 5] = acc;
  __syncthreads();
  if (threadIdx.x == 0) {
    float s = 0.0f;
#pragma unroll
    for (int k = 0; k < 8; ++k) s += red[k];
    out[1 + b] = s;
  }
}

// ---------------------------------------------------------------------------
// Kernel 2: parallel reduction over the tag axis (T=64), tag 0 masked out.
// One wave32 per (b,s) tile of shape W=16 x T=64 (4 KB).
// Produces  X[s][b][w] = exp( logsumexp_{t!=0} logits[b,s,w,t] )
//                      = exp(m_w) * sum_t exp(logit - m_w)
// The exp-sum over T is 16 accumulating V_WMMA_F32_16X16X4_F32 ops against an
// all-ones B matrix (row sums land in every column of C).
// ---------------------------------------------------------------------------
__global__ __launch_bounds__(128) void crf_rowlse(const float* __restrict__ inp,
                                                  float* __restrict__ X) {
  const int wave = threadIdx.x >> 5;
  const int lane = threadIdx.x & 31;
  const int tile = blockIdx.x * 4 + wave;        // 0 .. B*S-1, tile = b*S + s
  const int b = tile >> 9;
  const int s = tile & (Ssz - 1);

  const float* tp = inp + (size_t)tile * (Wsz * Tsz);
  const int row  = lane & 15;                    // matrix row M (width w)
  const int half = lane >> 4;                    // 0: K={0,1}  1: K={2,3}

  // Load directly in WMMA f32 16x16x4 A-matrix layout:
  // chunk c covers K = 4c..4c+3; this lane holds K = 4c + 2*half + {0,1}.
  v2f a[16];
#pragma unroll
  for (int c = 0; c < 16; ++c) {
    const float* p = tp + row * Tsz + 4 * c + 2 * half;
    a[c] = *(const v2f*)p;                       // global_load_b64
  }

  // Per-row max (include t=0: harmless, m only needs to be >= true max).
  float m = NEG_INF;
#pragma unroll
  for (int c = 0; c < 16; ++c) m = fmaxf(m, fmaxf(a[c].x, a[c].y));
  m = fmaxf(m, __shfl_xor(m, 16));               // combine the two K-halves

  // exp(x - m); zero out the DEFAULT_TAG (t == 0) contribution.
#pragma unroll
  for (int c = 0; c < 16; ++c) {
    a[c].x = __expf(a[c].x - m);
    a[c].y = __expf(a[c].y - m);
  }
  if (half == 0) a[0].x = 0.0f;                  // t=0 lives in lanes 0-15, chunk 0, .x

  // Row sums via WMMA: D = A x Ones + C  ->  every column n holds sum_k A[m,k].
  const v2f ones = {1.0f, 1.0f};
  v8f acc = {};
#pragma unroll
  for (int c = 0; c < 16; ++c) {
    acc = __builtin_amdgcn_wmma_f32_16x16x4_f32(
        /*neg_a=*/false, a[c], /*neg_b=*/false, ones,
        /*c_mod=*/(short)0, acc, /*reuse_a=*/false, /*reuse_b=*/false);
  }

  // C/D layout: lane L holds column N=L&15; rows 0-7 (L<16) / 8-15 (L>=16) in
  // elements 0-7.  Lane L also holds the max m of row M=L&15.  So lanes 0-7
  // and 24-31 hold BOTH the max and the column-sum of "their" row M, in
  // element M&7.  Extract via a select chain, scale, and store coalesced.
  const int M = lane & 15;
  const int e = M & 7;
  float sumv = acc[0];
  sumv = (e == 1) ? acc[1] : sumv;
  sumv = (e == 2) ? acc[2] : sumv;
  sumv = (e == 3) ? acc[3] : sumv;
  sumv = (e == 4) ? acc[4] : sumv;
  sumv = (e == 5) ? acc[5] : sumv;
  sumv = (e == 6) ? acc[6] : sumv;
  sumv = (e == 7) ? acc[7] : sumv;
  const float xval = __expf(m) * sumv;           // expE for row M
  if (lane < 8 || lane >= 24) {
    X[(size_t)s * (Bsz * Wsz) + b * Wsz + M] = xval;
  }
}

// ---------------------------------------------------------------------------
// Kernel 3: sequential semi-Markov forward scan over S=512 on the reduced
// X = exp(E) tensor (512 KB, L2-resident).  One wave; lane b < 16 owns batch
// row b with the ENTIRE width-window in registers, held in rescaled exp
// domain:  p[w] = exp(window[w] - alpha_prev).  Each step:
//     sum   = dot(p, X[s,b,:])          (fma tree, depth 4)
//     alpha += log(sum)                 (off the critical path)
//     p'    = [1, p[0]/sum, ..., p[14]/sum]   (one v_rcp + 15 muls)
// No cross-lane ops, no LDS, no exp in the loop -> minimal serial latency.
// Finally combines with the numerator already in d_out[1..16]:
//   d_out[0] = mean_b( numerator[b] - partition[b] ).
// ---------------------------------------------------------------------------
__global__ __launch_bounds__(32) void crf_scan(const float* __restrict__ X,
                                               const int* __restrict__ mask,
                                               float* __restrict__ out) {
  const int lane = threadIdx.x;
  const int b = lane & 15;
  __shared__ float spart[Bsz];

  if (lane < 16) {
    // length[b] = sum_s mask[b,s]
    int len = 0;
    const int* mb = mask + b * Ssz;
    for (int k = 0; k < Ssz; k += 4) {
      const int4 mm = *(const int4*)(mb + k);
      len += mm.x + mm.y + mm.z + mm.w;
    }

    float p[16];
    p[0] = 1.0f;
#pragma unroll
    for (int w = 1; w < 16; ++w) p[w] = 0.0f;
    float a = 0.0f;
    float part = 0.0f;

    // current row (registers), double-buffered against next row's load
    float xv[16];
    {
      const float* x0 = X + b * Wsz;
#pragma unroll
      for (int w = 0; w < 16; ++w) xv[w] = x0[w];
    }

    for (int s = 0; s < Ssz; ++s) {
      // Issue next row's loads before the dependent chain.
      float xn[16];
      if (s + 1 < Ssz) {
        const float* xp = X + (size_t)(s + 1) * (Bsz * Wsz) + b * Wsz;
#pragma unroll
        for (int w = 0; w < 16; ++w) xn[w] = xp[w];
      } else {
#pragma unroll
        for (int w = 0; w < 16; ++w) xn[w] = 0.0f;
      }
      if (s + 2 < Ssz)
        __builtin_prefetch(X + (size_t)(s + 2) * (Bsz * Wsz) + b * Wsz, 0, 1);

      // sum = dot(p, xv)  -- explicit tree to keep the chain at depth ~5
      float q[16];
#pragma unroll
      for (int w = 0; w < 16; ++w) q[w] = p[w] * xv[w];
#pragma unroll
      for (int st = 8; st >= 1; st >>= 1) {
#pragma unroll
        for (int w = 0; w < st; ++w) q[w] += q[w + st];
      }
      const float sum = q[0];

      a += __logf(sum);                 // trans pipe, off the critical path
      if (s == len - 1) part = a;

      const float r = __builtin_amdgcn_rcpf(sum);  // raw v_rcp_f32 (~1 ulp)
#pragma unroll
      for (int w = 15; w >= 1; --w) p[w] = p[w - 1] * r;
      p[0] = 1.0f;

#pragma unroll
      for (int w = 0; w < 16; ++w) xv[w] = xn[w];
    }
    spart[b] = part;
  }
  __syncthreads();
  if (lane == 0) {
    float acc = 0.0f;
#pragma unroll
    for (int bb = 0; bb < Bsz; ++bb) acc += out[1 + bb] - spart[bb];
    out[0] = acc / (float)Bsz;          // average over batches
  }
}

// ---------------------------------------------------------------------------
extern "C" void kernel_launch(void* const* d_in, const int* in_sizes, int n_in,
                              void* d_out, int out_size, void* d_ws, size_t ws_size,
                              hipStream_t stream) {
  (void)in_sizes; (void)n_in; (void)out_size; (void)ws_size;
  const float* inp  = (const float*)d_in[0];   // (B,S,W,T) f32
  const int*   tags = (const int*)d_in[1];     // (B,S,W)
  const int*   mask = (const int*)d_in[2];     // (B,S)
  float* out = (float*)d_out;                  // [loss, numerator[0..15]]
  float* X   = (float*)d_ws;                   // (S,B,W) f32 = 512 KB scratch

  // numerator first (crf_scan reads d_out[1..16]; same-stream ordering).
  crf_num<<<Bsz, 256, 0, stream>>>(inp, tags, mask, out);
  // parallel T-reduction via WMMA: 8192 tiles, 4 waves/block.
  crf_rowlse<<<(Bsz * Ssz) / 4, 128, 0, stream>>>(inp, X);
  // serial scan + final combine (one wave, 16 independent batch rows).
  crf_scan<<<1, 32, 0, stream>>>(X, mask, out);
}